// HF_2602750181657
// MI455X (gfx1250) — compile-verified
//
#include <hip/hip_runtime.h>
#include <cstdint>
#include <cstddef>

// Householder reflection per row: out = z - 2 * v * (v.z)/(v.v),  B=8192, L=4096, f32.
// Memory-bound (384 MB @ 23.3 TB/s ~= 16.5us floor). Strategy: one 256-thread block
// (8 wave32) per row; stage v-row and z-row (16KB each) into LDS with gfx1250 async
// global->LDS copies (ASYNCcnt), keep the data in registers for the epilogue so HBM
// sees exactly one read of v, one read of z, one NT write of out.

typedef __attribute__((ext_vector_type(4))) float f32x4;
typedef __attribute__((ext_vector_type(4))) int   i32x4;

// The async builtin wants typed v4i pointers: arg0 in addrspace(1) (global),
// arg1 in addrspace(3) (LDS).
#define AS1V4(p) ((__attribute__((address_space(1))) i32x4*)(p))
#define AS3V4(p) ((__attribute__((address_space(3))) i32x4*)(p))
#define AS3OFF(p) ((unsigned)(size_t)((__attribute__((address_space(3))) char*)(p)))

constexpr int kL          = 4096;          // row length in floats (fixed by reference)
constexpr int kTPB        = 256;           // 8 wave32 per block
constexpr int kWaves      = kTPB / 32;
constexpr int kF4         = kL / 4;        // 1024 float4 per row
constexpr int kF4PerThrd  = kF4 / kTPB;    // 4 float4 per thread

__launch_bounds__(kTPB)
__global__ void HF_householder_rows(const float* __restrict__ v,
                                    const float* __restrict__ z,
                                    float* __restrict__ out) {
    __shared__ f32x4 vsh[kF4];             // 16 KB
    __shared__ f32x4 zsh[kF4];             // 16 KB
    __shared__ float wvz[kWaves];
    __shared__ float wvv[kWaves];

    const int    tid  = threadIdx.x;
    const size_t base = (size_t)blockIdx.x * (size_t)kL;
    const f32x4* __restrict__ vrow = (const f32x4*)(v + base);
    const f32x4* __restrict__ zrow = (const f32x4*)(z + base);

    // ---- Stage both rows into LDS via CDNA5 async copy (ASYNCcnt, no VGPR bounce).
#if __has_builtin(__builtin_amdgcn_global_load_async_to_lds_b128)
#pragma unroll
    for (int j = 0; j < kF4PerThrd; ++j) {
        const int i4 = j * kTPB + tid;     // lane-consecutive 16B -> coalesced + bank-clean
        __builtin_amdgcn_global_load_async_to_lds_b128(AS1V4(vrow + i4), AS3V4(&vsh[i4]), 0, 0);
        __builtin_amdgcn_global_load_async_to_lds_b128(AS1V4(zrow + i4), AS3V4(&zsh[i4]), 0, 0);
    }
#else
#pragma unroll
    for (int j = 0; j < kF4PerThrd; ++j) {
        const int i4 = j * kTPB + tid;
        unsigned vdst = AS3OFF(&vsh[i4]);
        unsigned zdst = AS3OFF(&zsh[i4]);
        asm volatile("global_load_async_to_lds_b128 %0, %1, off"
                     :: "v"(vdst), "v"(vrow + i4) : "memory");
        asm volatile("global_load_async_to_lds_b128 %0, %1, off"
                     :: "v"(zdst), "v"(zrow + i4) : "memory");
    }
#endif

    // Wait for this wave's async copies, then workgroup barrier so all 8 waves'
    // LDS writes are visible (each thread reads back only the slots it copied).
#if __has_builtin(__builtin_amdgcn_s_wait_asynccnt)
    __builtin_amdgcn_s_wait_asynccnt(0);
#else
    asm volatile("s_wait_asynccnt 0" ::: "memory");
#endif
    __syncthreads();

    // ---- One LDS pass: capture data in registers, accumulate v.z and v.v partials.
    f32x4 vr[kF4PerThrd], zr[kF4PerThrd];
    float vz = 0.0f, vv = 0.0f;
#pragma unroll
    for (int j = 0; j < kF4PerThrd; ++j) {
        const int i4 = j * kTPB + tid;
        const f32x4 a = vsh[i4];
        const f32x4 b = zsh[i4];
        vr[j] = a;
        zr[j] = b;
        vz = fmaf(a.x, b.x, fmaf(a.y, b.y, fmaf(a.z, b.z, fmaf(a.w, b.w, vz))));
        vv = fmaf(a.x, a.x, fmaf(a.y, a.y, fmaf(a.z, a.z, fmaf(a.w, a.w, vv))));
    }

    // ---- wave32 butterfly reduction (5 steps; warpSize == 32 on gfx1250).
#pragma unroll
    for (int m = 16; m > 0; m >>= 1) {
        vz += __shfl_xor(vz, m, 32);
        vv += __shfl_xor(vv, m, 32);
    }
    if ((tid & 31) == 0) {
        wvz[tid >> 5] = vz;
        wvv[tid >> 5] = vv;
    }
    __syncthreads();

    // ---- Combine the 8 wave partials redundantly in every thread (8 adds, no 2nd barrier).
    float tvz = 0.0f, tvv = 0.0f;
#pragma unroll
    for (int w = 0; w < kWaves; ++w) {
        tvz += wvz[w];
        tvv += wvv[w];
    }
    const float s = 2.0f * tvz / tvv;

    // ---- Epilogue from registers; NT stores keep the write stream out of L2.
    f32x4* __restrict__ orow = (f32x4*)(out + base);
#pragma unroll
    for (int j = 0; j < kF4PerThrd; ++j) {
        const int i4 = j * kTPB + tid;
        const f32x4 a = vr[j];
        const f32x4 b = zr[j];
        f32x4 o;
        o.x = fmaf(-s, a.x, b.x);
        o.y = fmaf(-s, a.y, b.y);
        o.z = fmaf(-s, a.z, b.z);
        o.w = fmaf(-s, a.w, b.w);
        __builtin_nontemporal_store(o, orow + i4);
    }
}

extern "C" void kernel_launch(void* const* d_in, const int* in_sizes, int n_in,
                              void* d_out, int out_size, void* d_ws, size_t ws_size,
                              hipStream_t stream) {
    const float* v   = (const float*)d_in[0];
    const float* z   = (const float*)d_in[1];
    float*       out = (float*)d_out;

    const int B = in_sizes[0] / kL;   // 8192 rows; one block per row
    hipLaunchKernelGGL(HF_householder_rows, dim3(B), dim3(kTPB), 0, stream, v, z, out);
}